// SigmoidImplicitLayer_27582279975189
// MI455X (gfx1250) — compile-verified
//
#include <hip/hip_runtime.h>
#include <math.h>

// Fixed-point solve of z = tanh(z @ W^T + x) on MI455X (gfx1250).
// Picard iteration on WMMA f32 replaces the reference's batched-LU Newton:
// same fixed point, mapped onto v_wmma_f32_16x16x4_f32 + v_tanh_f32.

typedef __attribute__((ext_vector_type(2))) float v2f;
typedef __attribute__((ext_vector_type(8))) float v8f;

#define NN     128   // feature dim
#define LDSTR  132   // padded LDS row stride (dwords): bank-conflict-free, keeps 16B align
#define ROWS   16    // rows of z per block
#define ITERS  128   // Picard steps (contraction ~0.6 -> ample margin vs 1e-3)

#if defined(__has_builtin)
#if __has_builtin(__builtin_amdgcn_tanhf)
#define FAST_TANH(v) __builtin_amdgcn_tanhf(v)   // gfx1250 V_TANH_F32 (TRANS unit)
#endif
#endif
#ifndef FAST_TANH
#define FAST_TANH(v) tanhf(v)
#endif

__global__ __launch_bounds__(256, 1)
void fp_solve_kernel(const float* __restrict__ x,
                     const float* __restrict__ W,
                     float* __restrict__ out)
{
    __shared__ float sW[NN * LDSTR];    // W row-major, padded (66 KB)
    __shared__ float sZ[ROWS * LDSTR];  // current z rows, padded (8.25 KB)

    const int tid     = threadIdx.x;    // 0..255, 8 waves of 32
    const int wave    = tid >> 5;
    const int lane    = tid & 31;
    const int rowBase = blockIdx.x * ROWS;

    // Cooperative load of W into LDS as b128 (64 KB, read once per block).
    for (int i = tid; i < (NN * NN) / 4; i += 256) {
        int r = i >> 5;               // 32 float4 per W row
        int c = (i & 31) << 2;
        float4 w4 = ((const float4*)W)[i];
        *(float4*)&sW[r * LDSTR + c] = w4;
    }
    // z0 = tanh(x), matching the reference's initial iterate (b128 in/out).
    for (int i = tid; i < (ROWS * NN) / 4; i += 256) {
        int r = i >> 5;
        int c = (i & 31) << 2;
        float4 x4 = ((const float4*)x)[(size_t)rowBase * (NN / 4) + i];
        float4 z4;
        z4.x = FAST_TANH(x4.x);
        z4.y = FAST_TANH(x4.y);
        z4.z = FAST_TANH(x4.z);
        z4.w = FAST_TANH(x4.w);
        *(float4*)&sZ[r * LDSTR + c] = z4;
    }

    // Per-wave constants. Wave owns output columns [n0, n0+16).
    // C/D 16x16 f32 layout: VGPR r, lane L -> (M = r + 8*(L>=16), N = L&15).
    const int n0   = wave << 4;
    const int half = lane >> 4;       // 0: lanes 0-15, 1: lanes 16-31
    const int mh   = half << 3;       // +8 row offset for upper lane half
    const int nc   = lane & 15;       // tile column (C/D, B) / tile row (A)
    const int koff = half << 1;       // A/B K sub-offset per lane half

    // x tile for this wave, pre-laid-out in the C accumulator pattern.
    float xc[8];
#pragma unroll
    for (int r = 0; r < 8; ++r)
        xc[r] = x[(size_t)(rowBase + mh + r) * NN + n0 + nc];

    const int aBase = nc * LDSTR;          // A operand: row nc of z
    const int bBase = (n0 + nc) * LDSTR;   // B operand: W row n0+nc (B[k][n] = W[n0+n][k])

    __syncthreads();

    for (int it = 0; it < ITERS; ++it) {
        // Two independent accumulator chains for ILP; c0 seeded with x tile.
        v8f c0, c1;
#pragma unroll
        for (int r = 0; r < 8; ++r) { c0[r] = xc[r]; c1[r] = 0.0f; }

        // K = 128 as 32 v_wmma_f32_16x16x4_f32, fully unrolled, 2 chains.
#pragma unroll
        for (int k = 0; k < NN; k += 8) {
            const int kk = k + koff;
            v2f a0 = *(const v2f*)&sZ[aBase + kk];
            v2f b0 = *(const v2f*)&sW[bBase + kk];
            v2f a1 = *(const v2f*)&sZ[aBase + kk + 4];
            v2f b1 = *(const v2f*)&sW[bBase + kk + 4];
            // 8 args: (neg_a, A, neg_b, B, c_mod, C, reuse_a, reuse_b)
            c0 = __builtin_amdgcn_wmma_f32_16x16x4_f32(false, a0, false, b0,
                                                       (short)0, c0, false, false);
            c1 = __builtin_amdgcn_wmma_f32_16x16x4_f32(false, a1, false, b1,
                                                       (short)0, c1, false, false);
        }

        // z_new = tanh(z W^T + x): single TRANS op per element (v_tanh_f32),
        // no divergence, co-executes with other waves' WMMAs.
        float zn[8];
#pragma unroll
        for (int r = 0; r < 8; ++r) zn[r] = FAST_TANH(c0[r] + c1[r]);

        __syncthreads();   // all waves done reading sZ as A-operand
#pragma unroll
        for (int r = 0; r < 8; ++r)
            sZ[(mh + r) * LDSTR + n0 + nc] = zn[r];
        __syncthreads();   // z_new visible to all waves
    }

    // At convergence z == tanh(z W^T + x); the last sZ write is already one
    // application of the map, matching the reference's final step.
    for (int i = tid; i < (ROWS * NN) / 4; i += 256) {
        int r = i >> 5;
        int c = (i & 31) << 2;
        ((float4*)out)[(size_t)rowBase * (NN / 4) + i] = *(const float4*)&sZ[r * LDSTR + c];
    }
}

extern "C" void kernel_launch(void* const* d_in, const int* in_sizes, int n_in,
                              void* d_out, int out_size, void* d_ws, size_t ws_size,
                              hipStream_t stream) {
    const float* x = (const float*)d_in[0];   // [B, 128]
    const float* W = (const float*)d_in[1];   // [128, 128]
    float* out = (float*)d_out;               // [B, 128]
    const int B = in_sizes[0] / NN;           // 8192
    dim3 grid(B / ROWS);                      // 512 blocks
    fp_solve_kernel<<<grid, dim3(256), 0, stream>>>(x, W, out);
}